// PGAT_25091198943528
// MI455X (gfx1250) — compile-verified
//
#include <hip/hip_runtime.h>
#include <hip/hip_bf16.h>
#include <math.h>

// PGAT session-graph attention for MI455X (gfx1250, wave32, WMMA).
// Memory-bound (~215 MB traffic @ 23.3 TB/s ~ 10us floor); GEMMs in full fp32
// via V_WMMA_F32_16X16X4_F32. W matrices are staged in LDS PRE-PAIRED in the
// WMMA B-fragment layout so each fragment is one aligned ds_load_b64 (no
// v_mov repacking between DS returns and the WMMA operand pairs).

typedef __attribute__((ext_vector_type(2))) float v2f;
typedef __attribute__((ext_vector_type(8))) float v8f;

#define DIM 128
#define NTL 8 // 8 column tiles of 16 -> 128 cols

__device__ __forceinline__ v8f wmma_k4(v2f a, v2f b, v8f c) {
  // D(16x16,f32) = A(16x4,f32) * B(4x16,f32) + C
  return __builtin_amdgcn_wmma_f32_16x16x4_f32(
      /*neg_a=*/false, a, /*neg_b=*/false, b,
      /*c_mod=*/(short)0, c, /*reuse_a=*/false, /*reuse_b=*/false);
}

// Stage row-major W[DIM][DIM] into LDS with (k, k+1) rows interleaved per
// column:  sW[((k>>1)*DIM + c)*2 + (k&1)] = W[k][c]
// so a lane's B fragment (W[kk][col], W[kk+1][col]) is one aligned b64.
__device__ __forceinline__ void stage_W_paired(float* sW, const float* W,
                                               int tid) {
  const float4* Wv = (const float4*)W;
  for (int idx = tid; idx < DIM * DIM / 4; idx += 256) {
    const int k = idx >> 5;        // 32 float4 per row
    const int c = (idx & 31) << 2; // column of .x
    const float4 w = Wv[idx];
    float* dst = sW + (((k >> 1) * DIM) << 1) + (k & 1);
    dst[(c + 0) << 1] = w.x;
    dst[(c + 1) << 1] = w.y;
    dst[(c + 2) << 1] = w.z;
    dst[(c + 3) << 1] = w.w;
  }
}

// B fragment for k-step k0, column tile col base: pair index kp = k0/2 + half.
__device__ __forceinline__ v2f ldsB(const float* sW, int kp, int col) {
  return *(const v2f*)(sW + ((kp * DIM + col) << 1));
}

// ---------------------------------------------------------------------------
// Kernel 1: q[b,:] = U_feat[b,:]@W_user + b_user + feat_i[last_nodes[b],:]@W_last
// 256 threads = 8 waves; each wave owns a 16-row tile. One 64KB LDS buffer
// reused sequentially for the two weight matrices.
// ---------------------------------------------------------------------------
__global__ void __launch_bounds__(256)
k_project(const float* __restrict__ feat, const float* __restrict__ Ufeat,
          const float* __restrict__ Wuser, const float* __restrict__ buser,
          const float* __restrict__ Wlast, const int* __restrict__ lastn,
          float* __restrict__ q) {
  extern __shared__ float sW[]; // DIM*DIM floats = 64KB, paired layout
  const int lane = threadIdx.x & 31;
  const int wave = threadIdx.x >> 5;
  const int half = lane >> 4; // 0 or 1
  const int l15  = lane & 15;

  stage_W_paired(sW, Wuser, threadIdx.x);
  __syncthreads();

  const int tile = (blockIdx.x * 8 + wave) * 16;
  const int row  = tile + l15; // both lane halves cover rows 0..15
  const float* Ua = Ufeat + (size_t)row * DIM + half * 2;
  const int ln = lastn[row];
  const float* La = feat + (size_t)ln * DIM + half * 2;

  v8f acc[NTL];
#pragma unroll
  for (int nt = 0; nt < NTL; ++nt) {
    const float bv = buser[nt * 16 + l15];
#pragma unroll
    for (int j = 0; j < 8; ++j) acc[nt][j] = bv; // bias broadcast down rows
  }

  // GEMM 1: U_feat @ W_user
  for (int k0 = 0; k0 < DIM; k0 += 4) {
    v2f a = *(const v2f*)(Ua + k0);
    const int kp = (k0 >> 1) + half;
#pragma unroll
    for (int nt = 0; nt < NTL; ++nt)
      acc[nt] = wmma_k4(a, ldsB(sW, kp, nt * 16 + l15), acc[nt]);
  }

  __syncthreads(); // all waves done reading W_user
  stage_W_paired(sW, Wlast, threadIdx.x);
  __syncthreads();

  // GEMM 2: feat_i[last_nodes] @ W_last, accumulated into same C
  for (int k0 = 0; k0 < DIM; k0 += 4) {
    v2f a = *(const v2f*)(La + k0);
    const int kp = (k0 >> 1) + half;
#pragma unroll
    for (int nt = 0; nt < NTL; ++nt)
      acc[nt] = wmma_k4(a, ldsB(sW, kp, nt * 16 + l15), acc[nt]);
  }

  // C layout: VGPR v, lane l -> element [M = v + 8*(l>>4)][col = nt*16 + (l&15)]
#pragma unroll
  for (int nt = 0; nt < NTL; ++nt)
#pragma unroll
    for (int v = 0; v < 8; ++v)
      q[(size_t)(tile + v + 8 * half) * DIM + nt * 16 + l15] = acc[nt][v];
}

// ---------------------------------------------------------------------------
// Kernel 2: e[i] = sigmoid(q[seg[i],:] + feat_i[i,:]@W_key) . w_e + log(cnt[i])
// Fuses the big GEMM with the score reduction; feat_key never hits memory.
// ---------------------------------------------------------------------------
__global__ void __launch_bounds__(256)
k_scores(const float* __restrict__ feat, const float* __restrict__ Wkey,
         const float* __restrict__ q, const float* __restrict__ we_g,
         const float* __restrict__ cnt, const int* __restrict__ seg,
         float* __restrict__ e_out) {
  extern __shared__ float sW[]; // W_key, paired layout, 64KB
  const int lane = threadIdx.x & 31;
  const int wave = threadIdx.x >> 5;
  const int half = lane >> 4;
  const int l15  = lane & 15;

  stage_W_paired(sW, Wkey, threadIdx.x);
  __syncthreads();

  const int tile = (blockIdx.x * 8 + wave) * 16; // N = 1600*128, exact
  const float* Arow = feat + (size_t)(tile + l15) * DIM + half * 2;

  v8f acc[NTL];
#pragma unroll
  for (int nt = 0; nt < NTL; ++nt)
#pragma unroll
    for (int j = 0; j < 8; ++j) acc[nt][j] = 0.0f;

  for (int k0 = 0; k0 < DIM; k0 += 4) {
    v2f a = *(const v2f*)(Arow + k0); // streams the 16-row tile, 8B/lane
    const int kp = (k0 >> 1) + half;
#pragma unroll
    for (int nt = 0; nt < NTL; ++nt)
      acc[nt] = wmma_k4(a, ldsB(sW, kp, nt * 16 + l15), acc[nt]);
  }

  // Epilogue: per-row score. Lane l holds cols {nt*16 + l15}; w_e preloaded.
  float we[NTL];
#pragma unroll
  for (int nt = 0; nt < NTL; ++nt) we[nt] = we_g[nt * 16 + l15];

#pragma unroll
  for (int v = 0; v < 8; ++v) {
    const int row = tile + v + 8 * half;
    const int sg  = seg[row];
    const float* qrow = q + (size_t)sg * DIM;
    float rs = 0.0f;
#pragma unroll
    for (int nt = 0; nt < NTL; ++nt) {
      const float x = acc[nt][v] + qrow[nt * 16 + l15];
      const float s = 1.0f / (1.0f + __expf(-x)); // sigmoid
      rs += s * we[nt];
    }
    // reduce across the 16 lanes sharing this row (xor within lane[3:0])
    rs += __shfl_xor(rs, 1);
    rs += __shfl_xor(rs, 2);
    rs += __shfl_xor(rs, 4);
    rs += __shfl_xor(rs, 8);
    if (l15 == 0) e_out[row] = rs + __logf(cnt[row]);
  }
}

// ---------------------------------------------------------------------------
// Kernel 3: segment start offsets (segment_ids is sorted).
// start[b] = first i with seg[i] >= b; start[B] = N.
// ---------------------------------------------------------------------------
__global__ void k_bounds(const int* __restrict__ seg, int* __restrict__ start,
                         int N, int B) {
  const int i = blockIdx.x * blockDim.x + threadIdx.x;
  if (i > N) return;
  const int cur  = (i < N) ? seg[i] : B;
  const int prev = (i == 0) ? -1 : seg[i - 1];
  for (int b = prev + 1; b <= cur; ++b) start[b] = i;
}

// ---------------------------------------------------------------------------
// Kernel 4: per-segment softmax + weighted reduce of raw features.
// One 128-thread workgroup per segment (avg ~50 nodes). No atomics.
// rst[b,d] = sum_i exp(e_i - m) * feat[i,d] / sum_i exp(e_i - m)
// Coefficients for each 128-node chunk staged once in LDS (one exp per node),
// then broadcast-read while streaming coalesced feat columns.
// ---------------------------------------------------------------------------
__global__ void __launch_bounds__(128)
k_segment(const float* __restrict__ e, const float* __restrict__ feat,
          const int* __restrict__ start, float* __restrict__ rst) {
  __shared__ float red[4];
  __shared__ float mS, sS;
  __shared__ float coef[128];
  const int b   = blockIdx.x;
  const int tid = threadIdx.x;
  const int s0 = start[b];
  const int s1 = start[b + 1];

  // pass 1: segment max
  float lm = -3.402823466e38f;
  for (int i = s0 + tid; i < s1; i += 128) lm = fmaxf(lm, e[i]);
#pragma unroll
  for (int o = 16; o > 0; o >>= 1) lm = fmaxf(lm, __shfl_xor(lm, o));
  if ((tid & 31) == 0) red[tid >> 5] = lm;
  __syncthreads();
  if (tid == 0) mS = fmaxf(fmaxf(red[0], red[1]), fmaxf(red[2], red[3]));
  __syncthreads();
  const float m = mS;

  // pass 2: sum of exp
  float ls = 0.0f;
  for (int i = s0 + tid; i < s1; i += 128) ls += __expf(e[i] - m);
#pragma unroll
  for (int o = 16; o > 0; o >>= 1) ls += __shfl_xor(ls, o);
  if ((tid & 31) == 0) red[tid >> 5] = ls;
  __syncthreads();
  if (tid == 0) sS = red[0] + red[1] + red[2] + red[3];
  __syncthreads();
  const float inv = 1.0f / sS;

  // pass 3: thread tid owns column d = tid; chunked LDS-broadcast coefs.
  float acc = 0.0f;
  for (int base = s0; base < s1; base += 128) {
    const int cn = min(128, s1 - base);
    __syncthreads();
    if (tid < cn) coef[tid] = __expf(e[base + tid] - m) * inv;
    __syncthreads();
    for (int j = 0; j < cn; ++j)
      acc += coef[j] * feat[(size_t)(base + j) * DIM + tid];
  }
  rst[(size_t)b * DIM + tid] = acc;
}

// ---------------------------------------------------------------------------
extern "C" void kernel_launch(void* const* d_in, const int* in_sizes, int n_in,
                              void* d_out, int out_size, void* d_ws, size_t ws_size,
                              hipStream_t stream) {
  const float* feat  = (const float*)d_in[0]; // [N, D]
  const float* Ufeat = (const float*)d_in[1]; // [B, D]
  const float* cnt   = (const float*)d_in[2]; // [N]
  const float* Wkey  = (const float*)d_in[3]; // [D, D]
  const float* Wuser = (const float*)d_in[4]; // [D, D]
  const float* buser = (const float*)d_in[5]; // [D]
  const float* Wlast = (const float*)d_in[6]; // [D, D]
  const float* we    = (const float*)d_in[7]; // [D, 1]
  const int*   lastn = (const int*)d_in[8];   // [B]
  const int*   seg   = (const int*)d_in[9];   // [N]

  const int N = in_sizes[0] / DIM; // 204800 (divisible by 128)
  const int B = in_sizes[1] / DIM; // 4096  (divisible by 128)

  // workspace layout
  float* q     = (float*)d_ws;                // B*D floats
  float* e     = q + (size_t)B * DIM;         // N floats
  int*   start = (int*)(e + N);               // B+1 ints
  float* rst   = (float*)d_out;               // [B, D]

  const size_t lds = (size_t)DIM * DIM * sizeof(float); // 64KB

  k_project<<<B / 128, 256, lds, stream>>>(feat, Ufeat, Wuser, buser, Wlast,
                                           lastn, q);
  k_scores<<<N / 128, 256, lds, stream>>>(feat, Wkey, q, we, cnt, seg, e);
  k_bounds<<<(N + 1 + 255) / 256, 256, 0, stream>>>(seg, start, N, B);
  k_segment<<<B, 128, 0, stream>>>(e, feat, start, rst);
}